// Model_13254269076016
// MI455X (gfx1250) — compile-verified
//
#include <hip/hip_runtime.h>

typedef __attribute__((ext_vector_type(2))) float v2f;
typedef __attribute__((ext_vector_type(8))) float v8f;

#define IS_N    256
#define NFACES  1024
#define INV_FAR 0.01f   // 1/FAR : zp < FAR  <=> iz > 1/FAR
#define INV_NEAR 10.0f  // 1/NEAR: zp > NEAR <=> iz < 1/NEAR
#define EPS_P   1e-9f

// ---------------------------------------------------------------------------
// Kernel 1: per-face edge-function coefficients (camera + perspective + setup)
// coef layout: [edge_fn e=0..3][face f][pair hi=0..1] of float2
//   hi=0 -> (A,B)  (coeffs of xs, ys)     = B-fragment VGPR0(K=0)/VGPR1(K=1)
//   hi=1 -> (C,0)  (constant, K=3 pad)    = B-fragment VGPR0(K=2)/VGPR1(K=3)
// ---------------------------------------------------------------------------
__global__ void setup_faces_kernel(const float* __restrict__ verts,
                                   const int*   __restrict__ faces,
                                   float2*      __restrict__ coef,
                                   float*       __restrict__ out) {
  const int f = blockIdx.x * blockDim.x + threadIdx.x;
  if (f == 0) out[0] = 0.0f;                 // zero the loss accumulator
  if (f >= NFACES) return;

  // Camera (ELEV=0, AZIM=90deg, CAM_DIST=2.732), matching the reference.
  const double el = 0.0, az = 1.5707963267948966, cd = 2.732;
  const float ex = (float)( cd * cos(el) * sin(az));
  const float ey = (float)( cd * sin(el));
  const float ez = (float)(-cd * cos(el) * cos(az));
  float zx = -ex, zy = -ey, zz = -ez;
  float zi = rsqrtf(zx*zx + zy*zy + zz*zz);
  zx *= zi; zy *= zi; zz *= zi;
  float xx = zz, xy = 0.0f, xz = -zx;        // cross(up=(0,1,0), z)
  float xi = rsqrtf(xx*xx + xy*xy + xz*xz);
  xx *= xi; xy *= xi; xz *= xi;
  float yx = zy*xz - zz*xy;                  // cross(z, x)
  float yy = zz*xx - zx*xz;
  float yz = zx*xy - zy*xx;
  float yi = rsqrtf(yx*yx + yy*yy + yz*yz);
  yx *= yi; yy *= yi; yz *= yi;
  const float width = 0.57735026918962576f;  // tan(30 deg)

  float sx[3], sy[3], sz[3];
  #pragma unroll
  for (int k = 0; k < 3; ++k) {
    const int vi = faces[f*3 + k];
    const float tx = verts[vi*3+0] - ex;
    const float ty = verts[vi*3+1] - ey;
    const float tz = verts[vi*3+2] - ez;
    const float cx = xx*tx + xy*ty + xz*tz;
    const float cy = yx*tx + yy*ty + yz*tz;
    const float cz = zx*tx + zy*ty + zz*tz;
    sx[k] = cx / (cz * width);
    sy[k] = cy / (cz * width);
    sz[k] = cz;
  }
  const float x0 = sx[0], x1 = sx[1], x2 = sx[2];
  const float y0 = sy[0], y1 = sy[1], y2 = sy[2];
  const float z0 = sz[0], z1 = sz[1], z2 = sz[2];

  const float denom = (y1-y2)*(x0-x2) + (x2-x1)*(y0-y2);
  const bool  valid = fabsf(denom) > EPS_P;
  const float d = valid ? denom : 1.0f;
  float A0 = (y1-y2)/d, B0 = (x2-x1)/d;
  float A1 = (y2-y0)/d, B1 = (x0-x2)/d;
  float C0 = -(A0*x2 + B0*y2);
  const float C1 = -(A1*x2 + B1*y2);
  const float A2 = -(A0+A1), B2 = -(B0+B1), C2 = 1.0f - C0 - C1;
  const float P  = A0/z0 + A1/z1 + A2/z2;
  const float Q  = B0/z0 + B1/z1 + B2/z2;
  const float Rc = C0/z0 + C1/z1 + C2/z2;
  if (!valid) { A0 = 0.0f; B0 = 0.0f; C0 = -1.0f; }   // w0 == -1 => never inside

  coef[(0*NFACES + f)*2 + 0] = make_float2(A0, B0);
  coef[(0*NFACES + f)*2 + 1] = make_float2(C0, 0.0f);
  coef[(1*NFACES + f)*2 + 0] = make_float2(A1, B1);
  coef[(1*NFACES + f)*2 + 1] = make_float2(C1, 0.0f);
  coef[(2*NFACES + f)*2 + 0] = make_float2(A2, B2);
  coef[(2*NFACES + f)*2 + 1] = make_float2(C2, 0.0f);
  coef[(3*NFACES + f)*2 + 0] = make_float2(P,  Q);
  coef[(3*NFACES + f)*2 + 1] = make_float2(Rc, 0.0f);
}

// ---------------------------------------------------------------------------
// Kernel 2: WMMA rasterizer. One wave32 per 16-pixel tile, all 1024 faces.
// D = A(16 pixels x [xs,ys,1,0]) * B(4 x 16 faces) via v_wmma_f32_16x16x4_f32.
// Coverage: covered <=> min(w0, w1, w2, iz-1/FAR, 1/NEAR-iz) > 0, and
// OR over faces == (max over faces of that min) > 0  ->  per-row tail is just
// v_min3_num_f32 x2 (+shared v_pk_add_f32) + v_max_num_f32; no cmp/bit ops
// in the hot loop.
// ---------------------------------------------------------------------------
__global__ void __launch_bounds__(256)
raster_kernel(const float2* __restrict__ coef,
              const float*  __restrict__ image_ref,
              float*        __restrict__ out) {
  __shared__ float2 s_coef[4 * NFACES * 2];   // 64 KB, whole coefficient table
  for (int k = threadIdx.x; k < 4*NFACES*2; k += 256)
    s_coef[k] = coef[k];
  __syncthreads();

  const int lane  = threadIdx.x & 31;
  const int wave  = threadIdx.x >> 5;
  const int tile  = blockIdx.x * 8 + wave;    // 4096 tiles total
  const int m     = lane & 15;                // row (pixel) / column (face) idx
  const int hi    = lane >> 4;                // half-wave select
  const int pbase = tile * 16;

  // A fragment: lanes 0-15 hold (xs,ys) of pixel m; lanes 16-31 hold (1,0).
  const int p = pbase + m;
  const int j = p & (IS_N - 1);
  const int i = p >> 8;
  const float xs = (2.0f*(float)j + 1.0f - (float)IS_N) / (float)IS_N;
  const float ys = (2.0f*(float)(IS_N - 1 - i) + 1.0f - (float)IS_N) / (float)IS_N;
  v2f a;
  a.x = hi ? 1.0f : xs;
  a.y = hi ? 0.0f : ys;
  const v8f zeroC = {0.0f,0.0f,0.0f,0.0f,0.0f,0.0f,0.0f,0.0f};

  float covf[8];                              // max over faces of the min-test
  #pragma unroll
  for (int r = 0; r < 8; ++r) covf[r] = -3.4e38f;

  #pragma unroll 2
  for (int g = 0; g < NFACES/16; ++g) {
    const int f = g*16 + m;
    float2 t;
    v2f b0, b1, b2, bz;
    t = s_coef[(0*NFACES + f)*2 + hi]; b0.x = t.x; b0.y = t.y;
    t = s_coef[(1*NFACES + f)*2 + hi]; b1.x = t.x; b1.y = t.y;
    t = s_coef[(2*NFACES + f)*2 + hi]; b2.x = t.x; b2.y = t.y;
    t = s_coef[(3*NFACES + f)*2 + hi]; bz.x = t.x; bz.y = t.y;

    v8f d0 = __builtin_amdgcn_wmma_f32_16x16x4_f32(false, a, false, b0, (short)0, zeroC, false, false);
    v8f d1 = __builtin_amdgcn_wmma_f32_16x16x4_f32(false, a, false, b1, (short)0, zeroC, false, false);
    v8f d2 = __builtin_amdgcn_wmma_f32_16x16x4_f32(false, a, false, b2, (short)0, zeroC, false, false);
    v8f dz = __builtin_amdgcn_wmma_f32_16x16x4_f32(false, a, false, bz, (short)0, zeroC, false, false);

    #pragma unroll
    for (int r = 0; r < 8; ++r) {
      const float iz = dz[r];
      const float m3 = fminf(fminf(d0[r], d1[r]), d2[r]);            // v_min3
      const float m5 = fminf(fminf(m3, iz - INV_FAR), INV_NEAR - iz); // pk_add + v_min3
      covf[r] = fmaxf(covf[r], m5);                                   // v_max
    }
  }

  // max-reduce across the 16 columns of each half-wave.
  #pragma unroll
  for (int r = 0; r < 8; ++r) {
    covf[r] = fmaxf(covf[r], __shfl_xor(covf[r], 1, 32));
    covf[r] = fmaxf(covf[r], __shfl_xor(covf[r], 2, 32));
    covf[r] = fmaxf(covf[r], __shfl_xor(covf[r], 4, 32));
    covf[r] = fmaxf(covf[r], __shfl_xor(covf[r], 8, 32));
  }

  // Lane 0 owns pixels 0..7 of the tile, lane 16 owns pixels 8..15.
  if (m == 0) {
    float s = 0.0f;
    #pragma unroll
    for (int r = 0; r < 8; ++r) {
      const float cvd = (covf[r] > 0.0f) ? 1.0f : 0.0f;
      const float df  = cvd - image_ref[pbase + 8*hi + r];
      s += df * df;
    }
    atomicAdd(out, s);
  }
}

// ---------------------------------------------------------------------------
extern "C" void kernel_launch(void* const* d_in, const int* in_sizes, int n_in,
                              void* d_out, int out_size, void* d_ws, size_t ws_size,
                              hipStream_t stream) {
  const float* verts     = (const float*)d_in[0];   // (1,512,3)  f32
  const float* image_ref = (const float*)d_in[1];   // (1,256,256) f32
  const int*   faces     = (const int*)  d_in[2];   // (1,1024,3) i32
  float*  out  = (float*)d_out;                     // scalar loss
  float2* coef = (float2*)d_ws;                     // 4*1024*2 float2 = 64 KB

  setup_faces_kernel<<<dim3(4),   dim3(256), 0, stream>>>(verts, faces, coef, out);
  raster_kernel     <<<dim3(512), dim3(256), 0, stream>>>(coef, image_ref, out);
}